// RegistrationLoss_25134148616624
// MI455X (gfx1250) — compile-verified
//
#include <hip/hip_runtime.h>
#include <hip/hip_bf16.h>

#define BINS 64
#define NBATCH 2
#define EPSF 1e-10f

// ---------------- capability probes (device pass only) ---------------------
#if defined(__HIP_DEVICE_COMPILE__) && __has_builtin(__builtin_amdgcn_wmma_f32_16x16x4_f32)
#define HAVE_WMMA 1
#else
#define HAVE_WMMA 0
#endif

#if defined(__HIP_DEVICE_COMPILE__) && __has_builtin(__builtin_amdgcn_global_load_async_to_lds_b128)
#define USE_ASYNC 1
#else
#define USE_ASYNC 0
#endif

#if defined(__HIP_DEVICE_COMPILE__) && __has_builtin(__builtin_amdgcn_s_wait_asynccnt)
#define WAIT_ASYNC(n) __builtin_amdgcn_s_wait_asynccnt(n)
#else
#define WAIT_ASYNC(n) asm volatile("s_wait_asynccnt %0" ::"i"(n) : "memory")
#endif

typedef __attribute__((ext_vector_type(2))) float v2f;
typedef __attribute__((ext_vector_type(8))) float v8f;
typedef __attribute__((ext_vector_type(4))) int v4i;

#define GAS __attribute__((address_space(1)))
#define LAS __attribute__((address_space(3)))

// Order-preserving float -> uint key (works for negatives too).
__device__ __forceinline__ unsigned fkey(float x) {
  unsigned u = __float_as_uint(x);
  return (u & 0x80000000u) ? ~u : (u | 0x80000000u);
}
__device__ __forceinline__ float unkey(unsigned k) {
  unsigned u = (k & 0x80000000u) ? (k & 0x7FFFFFFFu) : ~k;
  return __uint_as_float(u);
}
__device__ __forceinline__ unsigned umn(unsigned a, unsigned b) { return a < b ? a : b; }
__device__ __forceinline__ unsigned umx(unsigned a, unsigned b) { return a > b ? a : b; }

// ---------------- kernel 0: init workspace ---------------------------------
__global__ void __launch_bounds__(256) init_kernel(unsigned* mm, unsigned* hist) {
  int i = blockIdx.x * blockDim.x + threadIdx.x;
  if (i < 4 * NBATCH) mm[i] = (i & 1) ? 0u : 0xFFFFFFFFu;  // even: min key, odd: max key
  if (i < NBATCH * BINS * BINS) hist[i] = 0u;
}

// ---------------- kernel 1: per-batch min/max of fixed & warped ------------
__global__ void __launch_bounds__(256) minmax_kernel(const float* __restrict__ f,
                                                     const float* __restrict__ w,
                                                     unsigned* __restrict__ mm, int n4) {
  const int b = blockIdx.y;
  const float4* f4 = (const float4*)f + (size_t)b * (size_t)n4;
  const float4* w4 = (const float4*)w + (size_t)b * (size_t)n4;
  unsigned fmn = 0xFFFFFFFFu, fmx = 0u, wmn = 0xFFFFFFFFu, wmx = 0u;
  for (int i = blockIdx.x * blockDim.x + threadIdx.x; i < n4; i += gridDim.x * blockDim.x) {
    float4 a = f4[i];
    float4 c = w4[i];
#define UPD(x, y)                               \
  {                                             \
    unsigned k = fkey(x);                       \
    fmn = umn(fmn, k); fmx = umx(fmx, k);       \
    unsigned q = fkey(y);                       \
    wmn = umn(wmn, q); wmx = umx(wmx, q);       \
  }
    UPD(a.x, c.x) UPD(a.y, c.y) UPD(a.z, c.z) UPD(a.w, c.w)
#undef UPD
  }
  __shared__ unsigned s[4];
  if (threadIdx.x == 0) { s[0] = 0xFFFFFFFFu; s[1] = 0u; s[2] = 0xFFFFFFFFu; s[3] = 0u; }
  __syncthreads();
  atomicMin(&s[0], fmn); atomicMax(&s[1], fmx);
  atomicMin(&s[2], wmn); atomicMax(&s[3], wmx);
  __syncthreads();
  if (threadIdx.x == 0) {
    atomicMin(&mm[b * 4 + 0], s[0]); atomicMax(&mm[b * 4 + 1], s[1]);
    atomicMin(&mm[b * 4 + 2], s[2]); atomicMax(&mm[b * 4 + 3], s[3]);
  }
}

// ---------------- kernel 2: joint histogram (LDS-privatized) ---------------
// Async double-buffered global->LDS staging when available; each thread owns
// its own 16B slot in each buffer, so no cross-wave sync is needed beyond the
// per-wave s_wait_asynccnt.
__global__ void __launch_bounds__(256) hist_kernel(const float* __restrict__ f,
                                                   const float* __restrict__ w,
                                                   const unsigned* __restrict__ mm,
                                                   unsigned* __restrict__ ghist, int n4) {
  const int b = blockIdx.y;
  __shared__ unsigned h[BINS * BINS];
  for (int i = threadIdx.x; i < BINS * BINS; i += blockDim.x) h[i] = 0u;

  const float fmin = unkey(mm[b * 4 + 0]);
  const float fmax = unkey(mm[b * 4 + 1]);
  const float wmin = unkey(mm[b * 4 + 2]);
  const float wmax = unkey(mm[b * 4 + 3]);
  const float fsc = (float)(BINS - 1) / (fmax - fmin + EPSF);
  const float wsc = (float)(BINS - 1) / (wmax - wmin + EPSF);
  __syncthreads();

  const float4* __restrict__ f4 = (const float4*)f + (size_t)b * (size_t)n4;
  const float4* __restrict__ w4 = (const float4*)w + (size_t)b * (size_t)n4;

  auto binpair = [&](float x, float y) {
    int fb = (int)((x - fmin) * fsc);
    fb = fb < 0 ? 0 : (fb > BINS - 1 ? BINS - 1 : fb);
    int wb = (int)((y - wmin) * wsc);
    wb = wb < 0 ? 0 : (wb > BINS - 1 ? BINS - 1 : wb);
    atomicAdd(&h[fb * BINS + wb], 1u);
  };
  auto consume = [&](float4 a, float4 c) {
    binpair(a.x, c.x); binpair(a.y, c.y); binpair(a.z, c.z); binpair(a.w, c.w);
  };

#if USE_ASYNC
  __shared__ float4 buf[2][2][256];  // [buffer][f/w][thread slot]  = 16 KB
  const int slot = threadIdx.x;
  const int tiles = n4 / 256;
  const int stride = gridDim.x;
  auto issue = [&](int t, int bi) {
    __builtin_amdgcn_global_load_async_to_lds_b128(
        (GAS v4i*)(f4 + (size_t)t * 256 + slot),
        (LAS v4i*)&buf[bi][0][slot], 0, 0);
    __builtin_amdgcn_global_load_async_to_lds_b128(
        (GAS v4i*)(w4 + (size_t)t * 256 + slot),
        (LAS v4i*)&buf[bi][1][slot], 0, 0);
  };
  int t = blockIdx.x;
  int cur = 0;
  if (t < tiles) issue(t, cur);
  for (; t < tiles; t += stride) {
    const int nt = t + stride;
    if (nt < tiles) {
      issue(nt, cur ^ 1);
      WAIT_ASYNC(2);  // the 2 just-issued may be outstanding; current tile done
    } else {
      WAIT_ASYNC(0);
    }
    consume(buf[cur][0][slot], buf[cur][1][slot]);
    cur ^= 1;
  }
  // tail (n4 not a multiple of 256)
  for (int i = tiles * 256 + blockIdx.x * blockDim.x + threadIdx.x; i < n4;
       i += gridDim.x * blockDim.x)
    consume(f4[i], w4[i]);
#else
  for (int i = blockIdx.x * blockDim.x + threadIdx.x; i < n4; i += gridDim.x * blockDim.x)
    consume(f4[i], w4[i]);
#endif

  __syncthreads();
  for (int i = threadIdx.x; i < BINS * BINS; i += blockDim.x) {
    unsigned v = h[i];
    if (v) atomicAdd(&ghist[b * BINS * BINS + i], v);
  }
}

// ---------------- kernel 3: entropies + NMI + affine MSE -------------------
// One block, 256 threads = 8 wave32s. Marginal row/col sums of the 64x64
// joint are computed with V_WMMA_F32_16X16X4_F32 (matrix x ones), exact in f32.
__global__ void __launch_bounds__(256) finalize_kernel(const float* __restrict__ pa,
                                                       const float* __restrict__ ta, int na,
                                                       const unsigned* __restrict__ ghist,
                                                       float* __restrict__ out) {
  __shared__ float J[NBATCH][BINS][BINS];  // 32 KB
  __shared__ float rs[NBATCH][BINS];
  __shared__ float cs[NBATCH][BINS];
  __shared__ float red0[256], red1[256];
  __shared__ float tot[2], hj[2], hf[2], hw[2];
  const int tid = threadIdx.x;
  float* Jf = &J[0][0][0];

  // counts -> shared, per-batch totals (deterministic tree reduction)
  float t0 = 0.f, t1 = 0.f;
  for (int i = tid; i < NBATCH * BINS * BINS; i += 256) {
    float c = (float)ghist[i];
    Jf[i] = c;
    if (i < BINS * BINS) t0 += c; else t1 += c;
  }
  red0[tid] = t0; red1[tid] = t1;
  __syncthreads();
  for (int s = 128; s > 0; s >>= 1) {
    if (tid < s) { red0[tid] += red0[tid + s]; red1[tid] += red1[tid + s]; }
    __syncthreads();
  }
  if (tid == 0) { tot[0] = red0[0]; tot[1] = red1[0]; }
  __syncthreads();

  // normalize + joint-entropy partial sums
  const float inv0 = 1.0f / (tot[0] + EPSF);
  const float inv1 = 1.0f / (tot[1] + EPSF);
  float hj0 = 0.f, hj1 = 0.f;
  for (int i = tid; i < NBATCH * BINS * BINS; i += 256) {
    float p = Jf[i] * ((i < BINS * BINS) ? inv0 : inv1);
    Jf[i] = p;
    float term = p * logf(p + EPSF);
    if (i < BINS * BINS) hj0 -= term; else hj1 -= term;
  }
  red0[tid] = hj0; red1[tid] = hj1;
  __syncthreads();
  for (int s = 128; s > 0; s >>= 1) {
    if (tid < s) { red0[tid] += red0[tid + s]; red1[tid] += red1[tid + s]; }
    __syncthreads();
  }
  if (tid == 0) { hj[0] = red0[0]; hj[1] = red1[0]; }
  __syncthreads();

  // ---- marginals: 8 waves, each owns (batch, 16-row block) ----
  const int wave = tid >> 5, lane = tid & 31;
  const int bb = wave >> 2, r0 = (wave & 3) * 16;
  const int m = lane & 15, hi = lane >> 4;
#if HAVE_WMMA
  {
    v2f ones; ones.x = 1.0f; ones.y = 1.0f;
    // row sums: D = sum_k A_k(16x4) * ones(4x16)
    v8f acc = {0.f, 0.f, 0.f, 0.f, 0.f, 0.f, 0.f, 0.f};
    for (int k0 = 0; k0 < BINS; k0 += 4) {
      const int kk = k0 + hi * 2;  // A 16x4 f32 layout: lanes16-31 hold K=2,3
      v2f a; a.x = J[bb][r0 + m][kk]; a.y = J[bb][r0 + m][kk + 1];
      acc = __builtin_amdgcn_wmma_f32_16x16x4_f32(false, a, false, ones, (short)0, acc,
                                                  false, false);
    }
    if (m == 0)
      for (int g = 0; g < 8; ++g) rs[bb][r0 + hi * 8 + g] = acc[g];
    // col sums = row sums of transpose
    v8f acc2 = {0.f, 0.f, 0.f, 0.f, 0.f, 0.f, 0.f, 0.f};
    for (int k0 = 0; k0 < BINS; k0 += 4) {
      const int kk = k0 + hi * 2;
      v2f a; a.x = J[bb][kk][r0 + m]; a.y = J[bb][kk + 1][r0 + m];
      acc2 = __builtin_amdgcn_wmma_f32_16x16x4_f32(false, a, false, ones, (short)0, acc2,
                                                   false, false);
    }
    if (m == 0)
      for (int g = 0; g < 8; ++g) cs[bb][r0 + hi * 8 + g] = acc2[g];
  }
#else
  if (tid < NBATCH * BINS) {
    const int b2 = tid >> 6, i2 = tid & 63;
    float r = 0.f, c = 0.f;
    for (int j = 0; j < BINS; ++j) { r += J[b2][i2][j]; c += J[b2][j][i2]; }
    rs[b2][i2] = r; cs[b2][i2] = c;
  }
#endif
  __syncthreads();

  // marginal entropies (tiny; serial per batch => deterministic)
  if (tid < NBATCH) {
    float a = 0.f, c = 0.f;
    for (int i = 0; i < BINS; ++i) {
      float p = rs[tid][i]; a -= p * logf(p + EPSF);
      float q = cs[tid][i]; c -= q * logf(q + EPSF);
    }
    hf[tid] = a; hw[tid] = c;
  }
  __syncthreads();

  if (tid == 0) {
    float mse = 0.f;
    for (int i = 0; i < na; ++i) { float d = pa[i] - ta[i]; mse += d * d; }
    mse /= (float)na;
    float nmis = 0.f;
    for (int b = 0; b < NBATCH; ++b) {
      float mi = hf[b] + hw[b] - hj[b];
      nmis += 2.0f * mi / (hf[b] + hw[b] + EPSF);
    }
    out[0] = mse - nmis / (float)NBATCH;
  }
}

// ---------------- launcher --------------------------------------------------
extern "C" void kernel_launch(void* const* d_in, const int* in_sizes, int n_in,
                              void* d_out, int out_size, void* d_ws, size_t ws_size,
                              hipStream_t stream) {
  (void)n_in; (void)out_size; (void)ws_size;
  const float* pa = (const float*)d_in[0];
  const float* ta = (const float*)d_in[1];
  const float* fx = (const float*)d_in[2];
  const float* wp = (const float*)d_in[3];
  float* out = (float*)d_out;

  const int na = in_sizes[0];            // 24
  const int n = in_sizes[2] / NBATCH;    // per-batch voxels (8,257,536)
  const int n4 = n / 4;                  // float4 count (2,064,384)

  unsigned* mm = (unsigned*)d_ws;        // 8 u32 min/max keys
  unsigned* hist = mm + 8;               // 2*64*64 u32 counts

  init_kernel<<<dim3(32), 256, 0, stream>>>(mm, hist);
  minmax_kernel<<<dim3(256, NBATCH), 256, 0, stream>>>(fx, wp, mm, n4);
  hist_kernel<<<dim3(252, NBATCH), 256, 0, stream>>>(fx, wp, mm, hist, n4);
  finalize_kernel<<<1, 256, 0, stream>>>(pa, ta, na, hist, out);
}